// MAGNN_Agg_9560597201168
// MI455X (gfx1250) — compile-verified
//
#include <hip/hip_runtime.h>

#define DIM 128

typedef __attribute__((ext_vector_type(2))) float v2f;
typedef __attribute__((ext_vector_type(8))) float v8f;

// ---------------------------------------------------------------------------
// sum[sidx[e]] += x[gidx[e]] * (ew ? ew[e] : 1);  cnt[sidx[e]] += 1
// One wave32 per edge; each lane handles 4 consecutive floats (float4 = 512B/row).
// ---------------------------------------------------------------------------
__global__ void edge_scatter_kernel(const float* __restrict__ x,
                                    const int* __restrict__ gidx,
                                    const int* __restrict__ sidx,
                                    const float* __restrict__ ew,
                                    float* __restrict__ sum,
                                    float* __restrict__ cnt,
                                    int E) {
  int wid  = (int)(((size_t)blockIdx.x * blockDim.x + threadIdx.x) >> 5);
  int lane = threadIdx.x & 31;
  if (wid >= E) return;
  int g = gidx[wid];
  int s = sidx[wid];
  float w = ew ? ew[wid] : 1.0f;
  const float4 v = ((const float4*)(x + (size_t)g * DIM))[lane];
  float* dst = sum + (size_t)s * DIM + lane * 4;
  __hip_atomic_fetch_add(dst + 0, v.x * w, __ATOMIC_RELAXED, __HIP_MEMORY_SCOPE_AGENT);
  __hip_atomic_fetch_add(dst + 1, v.y * w, __ATOMIC_RELAXED, __HIP_MEMORY_SCOPE_AGENT);
  __hip_atomic_fetch_add(dst + 2, v.z * w, __ATOMIC_RELAXED, __HIP_MEMORY_SCOPE_AGENT);
  __hip_atomic_fetch_add(dst + 3, v.w * w, __ATOMIC_RELAXED, __HIP_MEMORY_SCOPE_AGENT);
  if (lane == 0)
    __hip_atomic_fetch_add(cnt + s, 1.0f, __ATOMIC_RELAXED, __HIP_MEMORY_SCOPE_AGENT);
}

// out[i] = (sum[i]/max(cnt,1) + xadd[i]) * 0.5
__global__ void mean_combine_kernel(const float* __restrict__ sum,
                                    const float* __restrict__ cnt,
                                    const float* __restrict__ xadd,
                                    float* __restrict__ out, int total) {
  int i = blockIdx.x * blockDim.x + threadIdx.x;
  if (i >= total) return;
  float c = fmaxf(cnt[i >> 7], 1.0f);
  out[i] = (sum[i] / c + xadd[i]) * 0.5f;
}

// sum[i] /= max(cnt,1)  (in place)
__global__ void mean_inplace_kernel(float* __restrict__ sum,
                                    const float* __restrict__ cnt, int total) {
  int i = blockIdx.x * blockDim.x + threadIdx.x;
  if (i >= total) return;
  float c = fmaxf(cnt[i >> 7], 1.0f);
  sum[i] = sum[i] / c;
}

// ---------------------------------------------------------------------------
// out[n,j] = relu( sum_d feat[n,d]*W[j,d] + bias[j] )   (feat @ W.T + b)
// fp32 WMMA: V_WMMA_F32_16X16X4_F32. One wave per 16x16 tile; 8 waves/block
// cover all 128 columns for a 16-row stripe. K-loop = 32 wmma ops.
//
// A layout (16x4 f32): lane l holds A[M=l&15, K = 2*(l>>4) + {0,1}] in 2 VGPRs.
// B layout (4x16 f32): lane l holds B[K = 2*(l>>4) + {0,1}, N=l&15] = W[N, K..].
// C/D (16x16 f32):     vgpr r -> row (r + 8*(l>>4)), col l&15.
// ---------------------------------------------------------------------------
__global__ void gemm_bias_relu_kernel(const float* __restrict__ feat,
                                      const float* __restrict__ W,
                                      const float* __restrict__ bias,
                                      float* __restrict__ out, int n) {
  int lane   = threadIdx.x & 31;
  int waveId = threadIdx.x >> 5;        // 0..7 -> column tile
  int m0 = blockIdx.x * 16;
  int n0 = waveId * 16;

  int row = m0 + (lane & 15);
  if (row >= n) row = n - 1;            // tail-safe loads (stores predicated)
  int koff = (lane >> 4) << 1;          // 0 or 2
  int wrow = n0 + (lane & 15);          // output column = W row

  const float* fr = feat + (size_t)row * DIM + koff;
  const float* wr = W    + (size_t)wrow * DIM + koff;

  v8f c = {};
#pragma unroll
  for (int k = 0; k < DIM; k += 4) {
    v2f a = *(const v2f*)(fr + k);
    v2f b = *(const v2f*)(wr + k);
    c = __builtin_amdgcn_wmma_f32_16x16x4_f32(false, a, false, b,
                                              (short)0, c, false, false);
  }

  int col   = n0 + (lane & 15);
  float bv  = bias[col];
  int rbase = m0 + ((lane >> 4) << 3);
#pragma unroll
  for (int r = 0; r < 8; ++r) {
    int orow = rbase + r;
    if (orow < n)
      out[(size_t)orow * DIM + col] = fmaxf(c[r] + bv, 0.0f);
  }
}

// ---------------------------------------------------------------------------
// Attention fusion: logits[k]=dot(mp_k[n],att[k]); softmax over k; weighted sum.
// One wave32 per node. mp3 may alias out (read-before-write per element).
// ---------------------------------------------------------------------------
__global__ void attention_fuse_kernel(const float* __restrict__ mp0,
                                      const float* __restrict__ mp1,
                                      const float* __restrict__ mp2,
                                      const float* __restrict__ mp3,
                                      const float* __restrict__ att,
                                      float* __restrict__ out, int n) {
  int wid  = (int)(((size_t)blockIdx.x * blockDim.x + threadIdx.x) >> 5);
  int lane = threadIdx.x & 31;
  if (wid >= n) return;
  size_t base = (size_t)wid * DIM;

  float d0 = 0.f, d1 = 0.f, d2 = 0.f, d3 = 0.f;
#pragma unroll
  for (int d = lane; d < DIM; d += 32) {
    d0 += mp0[base + d] * att[d];
    d1 += mp1[base + d] * att[DIM + d];
    d2 += mp2[base + d] * att[2 * DIM + d];
    d3 += mp3[base + d] * att[3 * DIM + d];
  }
#pragma unroll
  for (int off = 16; off > 0; off >>= 1) {
    d0 += __shfl_down(d0, off, 32);
    d1 += __shfl_down(d1, off, 32);
    d2 += __shfl_down(d2, off, 32);
    d3 += __shfl_down(d3, off, 32);
  }
  d0 = __shfl(d0, 0, 32);
  d1 = __shfl(d1, 0, 32);
  d2 = __shfl(d2, 0, 32);
  d3 = __shfl(d3, 0, 32);

  float mx = fmaxf(fmaxf(d0, d1), fmaxf(d2, d3));
  float e0 = __expf(d0 - mx), e1 = __expf(d1 - mx);
  float e2 = __expf(d2 - mx), e3 = __expf(d3 - mx);
  float inv = 1.0f / (e0 + e1 + e2 + e3);
  e0 *= inv; e1 *= inv; e2 *= inv; e3 *= inv;

#pragma unroll
  for (int d = lane; d < DIM; d += 32) {
    float r = mp0[base + d] * e0 + mp1[base + d] * e1 +
              mp2[base + d] * e2 + mp3[base + d] * e3;
    out[base + d] = r;
  }
}

// ---------------------------------------------------------------------------
// Host-side helpers (all stream-ordered, graph-capture safe)
// ---------------------------------------------------------------------------
static void scatter_mean_pass(const float* x, const int* gi, const int* si,
                              const float* ew, float* sum, float* cnt,
                              int E, int nOut, hipStream_t stream) {
  hipMemsetAsync(sum, 0, (size_t)nOut * DIM * sizeof(float), stream);
  hipMemsetAsync(cnt, 0, (size_t)nOut * sizeof(float), stream);
  int blocks = (int)(((size_t)E * 32 + 255) / 256);
  edge_scatter_kernel<<<blocks, 256, 0, stream>>>(x, gi, si, ew, sum, cnt, E);
}

static void mean_combine(const float* sum, const float* cnt, const float* xadd,
                         float* out, int nOut, hipStream_t stream) {
  int total = nOut * DIM;
  mean_combine_kernel<<<(total + 255) / 256, 256, 0, stream>>>(sum, cnt, xadd, out, total);
}

static void mean_inplace(float* sum, const float* cnt, int nOut, hipStream_t stream) {
  int total = nOut * DIM;
  mean_inplace_kernel<<<(total + 255) / 256, 256, 0, stream>>>(sum, cnt, total);
}

static void gemm_bias_relu(const float* feat, const float* W, const float* bias,
                           float* out, int n, hipStream_t stream) {
  gemm_bias_relu_kernel<<<(n + 15) / 16, 256, 0, stream>>>(feat, W, bias, out, n);
}

extern "C" void kernel_launch(void* const* d_in, const int* in_sizes, int n_in,
                              void* d_out, int out_size, void* d_ws, size_t ws_size,
                              hipStream_t stream) {
  const float* x_node    = (const float*)d_in[0];
  const float* x0        = (const float*)d_in[1];
  const float* x2        = (const float*)d_in[2];
  const float* x3        = (const float*)d_in[3];
  const int*   ei_s1_src = (const int*)d_in[4];
  const int*   ei_s1_dst = (const int*)d_in[5];
  const int*   ei_s2_src = (const int*)d_in[6];
  const int*   ei_s2_dst = (const int*)d_in[7];
  const int*   ei12_a    = (const int*)d_in[8];
  const int*   ei12_b    = (const int*)d_in[9];
  const int*   ei13_a    = (const int*)d_in[10];
  const int*   ei13_b    = (const int*)d_in[11];
  const float* ew1       = (const float*)d_in[12];
  const float* ew2       = (const float*)d_in[13];
  const float* W_s1s     = (const float*)d_in[14];
  const float* b_s1s     = (const float*)d_in[15];
  const float* W_s2s     = (const float*)d_in[16];
  const float* b_s2s     = (const float*)d_in[17];
  const float* W_s121s   = (const float*)d_in[18];
  const float* b_s121s   = (const float*)d_in[19];
  const float* W_s131s   = (const float*)d_in[20];
  const float* b_s131s   = (const float*)d_in[21];
  const float* att_vec   = (const float*)d_in[22];

  const int N   = in_sizes[0] / DIM;
  const int N0  = in_sizes[1] / DIM;
  const int N2  = in_sizes[2] / DIM;
  const int N3  = in_sizes[3] / DIM;
  const int E1  = in_sizes[4];
  const int E2  = in_sizes[6];
  const int E12 = in_sizes[8];
  const int E13 = in_sizes[10];
  int Nm = N0 > N2 ? N0 : N2;
  if (N3 > Nm) Nm = N3;

  // Workspace layout (floats)
  float* ws   = (float*)d_ws;
  size_t ND   = (size_t)N * DIM;
  float* mp0  = ws;                         // s1s    [N,D]
  float* mp1  = mp0 + ND;                   // s2s    [N,D]
  float* mp2  = mp1 + ND;                   // s121s  [N,D]
  float* accN = mp2 + ND;                   // scatter accumulator over N
  float* cntN = accN + ND;                  // [N]
  float* net1 = cntN + N;                   // [N0,D] (reused by 3 metapaths)
  float* bufA = net1 + (size_t)N0 * DIM;    // [Nm,D]
  float* bufB = bufA + (size_t)Nm * DIM;    // [Nm,D]
  float* cntS = bufB + (size_t)Nm * DIM;    // [Nm]
  float* outf = (float*)d_out;              // doubles as mp3 (s131s)

  // ---- net1 = (scatter_mean(x_node[s1.src]*ew1, s1.dst, N0) + x0)*0.5
  scatter_mean_pass(x_node, ei_s1_src, ei_s1_dst, ew1, bufA, cntS, E1, N0, stream);
  mean_combine(bufA, cntS, x0, net1, N0, stream);

  // ---- s1s = relu(scatter_mean(net1[s1.dst], s1.src, N) @ W.T + b)
  scatter_mean_pass(net1, ei_s1_dst, ei_s1_src, nullptr, accN, cntN, E1, N, stream);
  mean_inplace(accN, cntN, N, stream);
  gemm_bias_relu(accN, W_s1s, b_s1s, mp0, N, stream);

  // ---- s2s
  scatter_mean_pass(x_node, ei_s2_src, ei_s2_dst, ew2, bufA, cntS, E2, N2, stream);
  mean_combine(bufA, cntS, x2, bufB, N2, stream);              // net2
  scatter_mean_pass(bufB, ei_s2_dst, ei_s2_src, nullptr, accN, cntN, E2, N, stream);
  mean_inplace(accN, cntN, N, stream);
  gemm_bias_relu(accN, W_s2s, b_s2s, mp1, N, stream);

  // ---- s121s
  scatter_mean_pass(net1, ei12_a, ei12_b, nullptr, bufA, cntS, E12, N2, stream);
  mean_combine(bufA, cntS, x2, bufB, N2, stream);              // n2
  scatter_mean_pass(bufB, ei12_b, ei12_a, nullptr, bufA, cntS, E12, N0, stream);
  mean_combine(bufA, cntS, x0, bufB, N0, stream);              // n3
  scatter_mean_pass(bufB, ei_s1_dst, ei_s1_src, ew1, accN, cntN, E1, N, stream);
  mean_inplace(accN, cntN, N, stream);
  gemm_bias_relu(accN, W_s121s, b_s121s, mp2, N, stream);

  // ---- s131s (into d_out, which also serves as mp3)
  scatter_mean_pass(net1, ei13_a, ei13_b, nullptr, bufA, cntS, E13, N3, stream);
  mean_combine(bufA, cntS, x3, bufB, N3, stream);              // n2b
  scatter_mean_pass(bufB, ei13_b, ei13_a, nullptr, bufA, cntS, E13, N0, stream);
  mean_combine(bufA, cntS, x0, bufB, N0, stream);              // n3b
  scatter_mean_pass(bufB, ei_s1_dst, ei_s1_src, ew1, accN, cntN, E1, N, stream);
  mean_inplace(accN, cntN, N, stream);
  gemm_bias_relu(accN, W_s131s, b_s131s, outf, N, stream);

  // ---- attention fusion over 4 metapaths
  attention_fuse_kernel<<<(N + 7) / 8, 256, 0, stream>>>(mp0, mp1, mp2, outf,
                                                         att_vec, outf, N);
}